// LinearAttention_66417374265873
// MI455X (gfx1250) — compile-verified
//
#include <hip/hip_runtime.h>
#include <hip/hip_bf16.h>
#include <math.h>

// ---------------------------------------------------------------------------
// LinearAttention on MI455X (gfx1250), fp32 path via V_WMMA_F32_16X16X4_F32,
// with double-buffered Tensor Data Mover (TDM) tile loads feeding the
// attention GEMMs. Memory-bound problem (77 GFLOP vs ~1.2 GB traffic).
// ---------------------------------------------------------------------------

#define B_    16
#define C_    256
#define NHW   4096      // 64*64 spatial
#define HEADS 8
#define DH    64
#define HD    512       // HEADS*DH
#define OC3   1536      // 3*HD

typedef __attribute__((ext_vector_type(2))) float        v2f;
typedef __attribute__((ext_vector_type(8))) float        v8f;
typedef __attribute__((ext_vector_type(4))) unsigned int v4u;
typedef __attribute__((ext_vector_type(8))) int          v8i;
typedef __attribute__((ext_vector_type(4))) int          v4i;

__device__ __forceinline__ v8f wmma_f32(v2f a, v2f b, v8f c) {
  // 8 args: (neg_a, A, neg_b, B, c_mod, C, reuse_a, reuse_b)
  return __builtin_amdgcn_wmma_f32_16x16x4_f32(false, a, false, b, (short)0, c,
                                               false, false);
}

// ---------------------------------------------------------------------------
// TDM: DMA one 64x64 fp32 tile (row stride `stride` elements) from global
// memory into LDS at byte offset lds_off, padding each 64-DWORD row by 2
// DWORDs -> LDS image float[64][66] (rows stay 8-byte aligned for ds b64).
// D# layout per CDNA5 ISA ch.8 (group0: count/lds/global/type, group1: dims).
// ---------------------------------------------------------------------------
__device__ __forceinline__ void tdm_load_tile64(unsigned lds_off,
                                                const float* gptr,
                                                unsigned stride) {
  const unsigned long long ga = (unsigned long long)(const void*)gptr;
  const v4u g0 = {
      1u,                                               // count=1, user mode
      lds_off,                                          // lds_addr (bytes)
      (unsigned)ga,                                     // global_addr[31:0]
      ((unsigned)(ga >> 32) & 0x1FFFFFFu) | 0x80000000u // addr[56:32] | type=2
  };
  const v8i g1 = {
      (int)((2u << 16) | (1u << 20) | (5u << 22) | (1u << 25)),
      //  data_size=4B | pad_enable | pad_interval=64dw | pad_amount=2dw
      (int)((stride & 0xFFFFu) << 16),                  // tensor_dim0 lo16
      (int)(((stride >> 16) & 0xFFFFu) | (64u << 16)),  // dim0 hi16 | dim1=64
      (int)(64u << 16),                                 // dim1 hi=0 | tile_dim0=64
      64,                                               // tile_dim1=64 | tile_dim2=0
      (int)stride,                                      // tensor_dim0_stride lo32
      0, 0                                              // stride hi / dim1_stride
  };
  const v4i g2 = {0, 0, 0, 0};
  const v4i g3 = {0, 0, 0, 0};
#if defined(__clang_major__) && (__clang_major__ >= 23)
  const v8i g4 = {0, 0, 0, 0, 0, 0, 0, 0};
  __builtin_amdgcn_tensor_load_to_lds(g0, g1, g2, g3, g4, 0);
#else
  __builtin_amdgcn_tensor_load_to_lds(g0, g1, g2, g3, 0);
#endif
}

__device__ __forceinline__ unsigned lds_off_of(const void* p) {
  return (unsigned)(unsigned long long)p;   // LDS aperture: offset = addr[31:0]
}

// ---------------------------------------------------------------------------
// K1: fused RMS-norm + QKV GEMM.
//   qkv[b][o][p] = (sum_c qkv_w[o][c] * norm_w[c] * x[b][c][p]) * s[b][p]
//   s[b][p] = 16 / max(||x[b][:][p]||, 1e-12)
// Workgroup = 128 threads (4 waves), 64(M) x 64(N) tile, K=256.
// x tile stored K-transposed ([col][k], 20-float rows) so each B fragment is
// one contiguous 8B pair -> single ds_load_b64, no register shuffling.
// ---------------------------------------------------------------------------
__global__ __launch_bounds__(128) void k_qkv(const float* __restrict__ x,
                                             const float* __restrict__ norm_w,
                                             const float* __restrict__ qkv_w,
                                             float* __restrict__ qkv) {
  const int nt = blockIdx.x;            // 0..63
  const int mt = blockIdx.y;            // 0..23
  const int b  = blockIdx.z;            // 0..15
  const int t  = threadIdx.x;           // 0..127
  const int lane = t & 31;
  const int w    = t >> 5;              // wave 0..3
  const int p0 = nt * 64;
  const int m0 = mt * 64;
  const float* xb = x + (size_t)b * C_ * NHW;

  __shared__ float xs[64][20];          // [col][k] K-transposed tile
  __shared__ float sqred[2][64];
  __shared__ float scol[64];

  v8f acc[4] = {};
  float sq = 0.f;
  const int lcol  = t & 63;
  const int lrow0 = (t >> 6) * 8;

  for (int k0 = 0; k0 < C_; k0 += 16) {
    if (k0 + 16 < C_)   // prefetch next k-tile (global_prefetch_b8 path)
      __builtin_prefetch(&xb[(size_t)(k0 + 16 + lrow0) * NHW + p0 + lcol], 0, 1);
    // stage x tile transposed; fold norm_w, accumulate raw sum of squares
#pragma unroll
    for (int i = 0; i < 8; ++i) {
      const int r = lrow0 + i;
      const float xv = xb[(size_t)(k0 + r) * NHW + p0 + lcol];
      sq += xv * xv;
      xs[lcol][r] = xv * norm_w[k0 + r];
    }
    __syncthreads();
#pragma unroll
    for (int kk = 0; kk < 16; kk += 4) {
      const int ksub = kk + ((lane >> 4) << 1);
      const int arow = m0 + w * 16 + (lane & 15);
      const v2f a = *(const v2f*)&qkv_w[(size_t)arow * C_ + k0 + ksub];
#pragma unroll
      for (int j = 0; j < 4; ++j) {
        const v2f bb = *(const v2f*)&xs[j * 16 + (lane & 15)][ksub];
        acc[j] = wmma_f32(a, bb, acc[j]);
      }
    }
    __syncthreads();
  }

  // reduce column sums of squares -> per-column scale
  sqred[t >> 6][lcol] = sq;
  __syncthreads();
  if (t < 64) {
    const float tot = sqred[0][t] + sqred[1][t];
    scol[t] = 16.0f / fmaxf(sqrtf(tot), 1e-12f);
  }
  __syncthreads();

  float* outb = qkv + (size_t)b * OC3 * NHW;
#pragma unroll
  for (int j = 0; j < 4; ++j) {
    const int ncol = j * 16 + (lane & 15);
    const float s = scol[ncol];
    const int mbase = m0 + w * 16 + ((lane >> 4) << 3);
#pragma unroll
    for (int r = 0; r < 8; ++r)
      outb[(size_t)(mbase + r) * NHW + p0 + ncol] = acc[j][r] * s;
  }
}

// ---------------------------------------------------------------------------
// K2a: per-row online-softmax stats for k (rows = b*512 + h*64 + d).
// One wave32 per row; shuffle reduction. Writes (max, 1/sumexp).
// ---------------------------------------------------------------------------
__global__ __launch_bounds__(256) void k_kstats(const float* __restrict__ qkv,
                                                float* __restrict__ kstats) {
  const int wid  = blockIdx.x * 8 + (threadIdx.x >> 5);   // 0..8191
  const int lane = threadIdx.x & 31;
  const int b    = wid >> 9;
  const int rem  = wid & 511;                              // h*64 + d
  const float* krow = qkv + (size_t)b * OC3 * NHW + (size_t)(HD + rem) * NHW;

  float m = -3.0e38f, s = 0.f;
  for (int i = lane; i < NHW; i += 32) {
    const float v = krow[i];
    const float nm = fmaxf(m, v);
    s = s * expf(m - nm) + expf(v - nm);
    m = nm;
  }
#pragma unroll
  for (int off = 16; off >= 1; off >>= 1) {
    const float om = __shfl_xor(m, off, 32);
    const float os = __shfl_xor(s, off, 32);
    const float nm = fmaxf(m, om);
    s = s * expf(m - nm) + os * expf(om - nm);
    m = nm;
  }
  if (lane == 0) {
    kstats[wid * 2]     = m;
    kstats[wid * 2 + 1] = 1.0f / s;
  }
}

// ---------------------------------------------------------------------------
// K2b: ctx[b][h][d][e] = sum_n softmax_n(k)[d][n] * v[e][n]
// One workgroup (8 waves) per (b,h); K=4096 streamed in 64-wide tiles that
// are DMA'd into LDS by the TDM, DOUBLE-BUFFERED: wave0 issues the next
// tile pair, then s_wait_tensorcnt 2 guarantees the current pair landed
// (TDM ops from one wave complete in order) while the next pair is in
// flight behind the WMMA work. k gets an in-LDS softmax transform.
// ---------------------------------------------------------------------------
__global__ __launch_bounds__(256) void k_ctx(const float* __restrict__ qkv,
                                             const float* __restrict__ kstats,
                                             float* __restrict__ ctx) {
  const int bh = blockIdx.x;            // 0..127
  const int b = bh >> 3, h = bh & 7;
  const int t = threadIdx.x, lane = t & 31, w = t >> 5;
  const float* kbase = qkv + (size_t)b * OC3 * NHW + (size_t)(HD + h * DH) * NHW;
  const float* vbase = qkv + (size_t)b * OC3 * NHW + (size_t)(2 * HD + h * DH) * NHW;

  __shared__ float ks[2][64][66];   // k tiles [d][n] (TDM image, 2-dword pad)
  __shared__ float vs[2][64][66];   // v tiles [e][n] (TDM image)
  __shared__ float kmx[64], kiv[64];

  if (t < 64) {
    kmx[t] = kstats[((bh << 6) + t) * 2];
    kiv[t] = kstats[((bh << 6) + t) * 2 + 1];
  }

  v8f acc[2] = {};
  const int mtl = w & 3;                // d-row tile (16 rows)
  const int j0  = (w >> 2) * 2;         // e-col tiles j0, j0+1
  const int lcol  = t & 63;
  const int lrow0 = (t >> 6) * 16;

  if (w == 0) {                         // prologue: DMA first tile pair
    tdm_load_tile64(lds_off_of(&ks[0][0][0]), kbase, NHW);
    tdm_load_tile64(lds_off_of(&vs[0][0][0]), vbase, NHW);
  }
  __syncthreads();

  for (int n0 = 0; n0 < NHW; n0 += 64) {
    const int p = (n0 >> 6) & 1;
    if (w == 0) {
      if (n0 + 64 < NHW) {              // DMA next pair into other buffer
        tdm_load_tile64(lds_off_of(&ks[p ^ 1][0][0]), kbase + n0 + 64, NHW);
        tdm_load_tile64(lds_off_of(&vs[p ^ 1][0][0]), vbase + n0 + 64, NHW);
        __builtin_amdgcn_s_wait_tensorcnt(2);   // current pair complete
      } else {
        __builtin_amdgcn_s_wait_tensorcnt(0);
      }
    }
    __syncthreads();

    // in-LDS softmax transform of the k tile
#pragma unroll
    for (int i = 0; i < 16; ++i) {
      const int r = lrow0 + i;
      ks[p][r][lcol] = expf(ks[p][r][lcol] - kmx[r]) * kiv[r];
    }
    __syncthreads();

#pragma unroll
    for (int kk = 0; kk < 64; kk += 4) {
      const int ksub = kk + ((lane >> 4) << 1);
      const v2f a = *(const v2f*)&ks[p][mtl * 16 + (lane & 15)][ksub];
#pragma unroll
      for (int jj = 0; jj < 2; ++jj) {
        const v2f bb = *(const v2f*)&vs[p][(j0 + jj) * 16 + (lane & 15)][ksub];
        acc[jj] = wmma_f32(a, bb, acc[jj]);
      }
    }
    __syncthreads();
  }

  float* cb = ctx + (size_t)bh * DH * DH;
#pragma unroll
  for (int jj = 0; jj < 2; ++jj) {
    const int e = (j0 + jj) * 16 + (lane & 15);
    const int dbase = mtl * 16 + ((lane >> 4) << 3);
#pragma unroll
    for (int r = 0; r < 8; ++r)
      cb[(dbase + r) * DH + e] = acc[jj][r];
  }
}

// ---------------------------------------------------------------------------
// K2c: attn[b][h*64+e][n] = sum_d ctx[d][e] * (softmax_d(q)*scale)[d][n]
// One workgroup per (b,h, 64-wide n tile). Both tiles DMA'd by the TDM:
// ctx tile (stride 64) and q tile (stride 4096), images [64][66].
// Column softmax of q done in LDS after the DMA lands.
// ---------------------------------------------------------------------------
__global__ __launch_bounds__(256) void k_attnout(const float* __restrict__ qkv,
                                                 const float* __restrict__ ctx,
                                                 float* __restrict__ attn) {
  const int nt = blockIdx.x;            // 0..63
  const int bh = blockIdx.y;            // 0..127
  const int b = bh >> 3, h = bh & 7;
  const int t = threadIdx.x, lane = t & 31, w = t >> 5;
  const int n0 = nt * 64;
  const float* qbase = qkv + (size_t)b * OC3 * NHW + (size_t)(h * DH) * NHW;
  const float* cb    = ctx + (size_t)bh * DH * DH;

  __shared__ float cs[64][66];  // TDM image: cs[d][e] = ctx[d][e]
  __shared__ float qs[64][66];  // TDM image: qs[d][n]

  if (w == 0) {
    tdm_load_tile64(lds_off_of(&cs[0][0]), cb, DH);
    tdm_load_tile64(lds_off_of(&qs[0][0]), qbase + n0, NHW);
    __builtin_amdgcn_s_wait_tensorcnt(0);
  }
  __syncthreads();

  // per-column softmax over d (64) + scale = DIM_HEAD^-0.5 = 1/8
  if (t < 64) {
    float mx = -3.0e38f;
    for (int d = 0; d < 64; ++d) mx = fmaxf(mx, qs[d][t]);
    float sum = 0.f;
    for (int d = 0; d < 64; ++d) { const float e = expf(qs[d][t] - mx); qs[d][t] = e; sum += e; }
    const float inv = 0.125f / sum;
    for (int d = 0; d < 64; ++d) qs[d][t] *= inv;
  }
  __syncthreads();

  v8f acc[2] = {};
  const int mtl = w & 3;                // e tile
  const int j0  = (w >> 2) * 2;         // n tiles
#pragma unroll
  for (int kk = 0; kk < 64; kk += 4) {
    const int ksub = kk + ((lane >> 4) << 1);
    v2f a;                              // A[m=e][k=d] = cs[d][e]
    a.x = cs[ksub][mtl * 16 + (lane & 15)];
    a.y = cs[ksub + 1][mtl * 16 + (lane & 15)];
#pragma unroll
    for (int jj = 0; jj < 2; ++jj) {
      v2f bb;                           // B[k=d][n] = qs[d][n]
      bb.x = qs[ksub][(j0 + jj) * 16 + (lane & 15)];
      bb.y = qs[ksub + 1][(j0 + jj) * 16 + (lane & 15)];
      acc[jj] = wmma_f32(a, bb, acc[jj]);
    }
  }

  float* ab = attn + (size_t)b * HD * NHW + (size_t)(h * DH) * NHW;
#pragma unroll
  for (int jj = 0; jj < 2; ++jj) {
    const int ncol = (j0 + jj) * 16 + (lane & 15);
    const int ebase = mtl * 16 + ((lane >> 4) << 3);
#pragma unroll
    for (int r = 0; r < 8; ++r)
      ab[(size_t)(ebase + r) * NHW + n0 + ncol] = acc[jj][r];
  }
}

// ---------------------------------------------------------------------------
// K3: out projection (M=256 full, K=512) + bias + final RMS-norm.
// 8 waves; wave w owns rows w*32..w*32+31 (2 m-tiles) x 4 n-tiles.
// attn tile stored K-transposed ([col][k], 20-float rows) for b64 fragments.
// Column sum-of-squares reduced through LDS atomics for the rms norm.
// ---------------------------------------------------------------------------
__global__ __launch_bounds__(256) void k_proj(const float* __restrict__ attn,
                                              const float* __restrict__ out_w,
                                              const float* __restrict__ out_b,
                                              const float* __restrict__ out_nw,
                                              float* __restrict__ out) {
  const int nt = blockIdx.x;            // 0..63
  const int b  = blockIdx.y;            // 0..15
  const int t = threadIdx.x, lane = t & 31, w = t >> 5;
  const int n0 = nt * 64;
  const float* ab = attn + (size_t)b * HD * NHW;

  __shared__ float as[64][20];          // [col][k] K-transposed tile
  __shared__ float biasw[256], gamw[256];
  __shared__ float colsq[64];

  if (t < 64) colsq[t] = 0.f;
  biasw[t] = out_b[t];
  gamw[t]  = out_nw[t];

  v8f acc[2][4] = {};
  const int lcol  = t & 63;
  const int krow0 = (t >> 6) * 4;

  for (int k0 = 0; k0 < HD; k0 += 16) {
    if (k0 + 16 < HD)
      __builtin_prefetch(&ab[(size_t)(k0 + 16 + krow0) * NHW + n0 + lcol], 0, 1);
#pragma unroll
    for (int i = 0; i < 4; ++i) {
      const int r = krow0 + i;
      as[lcol][r] = ab[(size_t)(k0 + r) * NHW + n0 + lcol];
    }
    __syncthreads();
#pragma unroll
    for (int kk = 0; kk < 16; kk += 4) {
      const int ksub = kk + ((lane >> 4) << 1);
      v2f bb[4];
#pragma unroll
      for (int j = 0; j < 4; ++j)
        bb[j] = *(const v2f*)&as[j * 16 + (lane & 15)][ksub];
#pragma unroll
      for (int mi = 0; mi < 2; ++mi) {
        const int mrow = w * 32 + mi * 16 + (lane & 15);
        const v2f a = *(const v2f*)&out_w[(size_t)mrow * HD + k0 + ksub];
#pragma unroll
        for (int j = 0; j < 4; ++j)
          acc[mi][j] = wmma_f32(a, bb[j], acc[mi][j]);
      }
    }
    __syncthreads();
  }

  // bias + column sum of squares
#pragma unroll
  for (int mi = 0; mi < 2; ++mi) {
    const int mbase = w * 32 + mi * 16 + ((lane >> 4) << 3);
#pragma unroll
    for (int j = 0; j < 4; ++j) {
      float part = 0.f;
#pragma unroll
      for (int r = 0; r < 8; ++r) {
        const float v = acc[mi][j][r] + biasw[mbase + r];
        acc[mi][j][r] = v;
        part += v * v;
      }
      atomicAdd(&colsq[j * 16 + (lane & 15)], part);
    }
  }
  __syncthreads();

  float* ob = out + (size_t)b * C_ * NHW;
#pragma unroll
  for (int j = 0; j < 4; ++j) {
    const int ncol = j * 16 + (lane & 15);
    const float s = 16.0f / fmaxf(sqrtf(colsq[ncol]), 1e-12f);
#pragma unroll
    for (int mi = 0; mi < 2; ++mi) {
      const int mbase = w * 32 + mi * 16 + ((lane >> 4) << 3);
#pragma unroll
      for (int r = 0; r < 8; ++r)
        ob[(size_t)(mbase + r) * NHW + n0 + ncol] = acc[mi][j][r] * s * gamw[mbase + r];
    }
  }
}

// ---------------------------------------------------------------------------
extern "C" void kernel_launch(void* const* d_in, const int* in_sizes, int n_in,
                              void* d_out, int out_size, void* d_ws, size_t ws_size,
                              hipStream_t stream) {
  const float* x       = (const float*)d_in[0];
  const float* norm_w  = (const float*)d_in[1];
  const float* qkv_w   = (const float*)d_in[2];
  const float* out_w   = (const float*)d_in[3];
  const float* out_b   = (const float*)d_in[4];
  const float* out_nw  = (const float*)d_in[5];
  float* out = (float*)d_out;

  float* wsf   = (float*)d_ws;
  float* qkv   = wsf;                                   // B*1536*4096
  float* kst   = qkv + (size_t)B_ * OC3 * NHW;          // B*8*64*2
  float* ctx   = kst + (size_t)B_ * HEADS * DH * 2;     // B*8*64*64
  float* attn  = ctx + (size_t)B_ * HEADS * DH * DH;    // B*512*4096

  k_qkv<<<dim3(64, 24, B_), 128, 0, stream>>>(x, norm_w, qkv_w, qkv);
  k_kstats<<<dim3(B_ * HEADS * DH / 8), 256, 0, stream>>>(qkv, kst);
  k_ctx<<<dim3(B_ * HEADS), 256, 0, stream>>>(qkv, kst, ctx);
  k_attnout<<<dim3(64, B_ * HEADS), 256, 0, stream>>>(qkv, ctx, attn);
  k_proj<<<dim3(64, B_), 256, 0, stream>>>(attn, out_w, out_b, out_nw, out);
}